// NeuralGDEForecaster_7035156431298
// MI455X (gfx1250) — compile-verified
//
#include <hip/hip_runtime.h>
#include <hip/hip_bf16.h>

// MI455X / gfx1250, wave32. f32 WMMA (V_WMMA_F32_16X16X4_F32) for all GEMMs:
// problem is latency-bound (~100 MFLOP, 2 MB traffic), so keep reference f32
// precision and use the matrix pipe for the 64-deep K reductions.

typedef float v2f __attribute__((ext_vector_type(2)));
typedef float v8f __attribute__((ext_vector_type(8)));

#define LDIM 68  // padded LDS row stride (floats): 8B-aligned b64 A-loads, conflict-free

__device__ __forceinline__ float sigmoidf_(float v) { return 1.0f / (1.0f + __expf(-v)); }

// One 16x16 f32 C-tile of  C = A(LDS, [Mx64], stride LDIM) @ W(global, [64xNcols], stride 64)
// via 16 chained v_wmma_f32_16x16x4_f32 (K = 16*4 = 64).
// f32 A 16x4 layout: VGPR0 = K0|K2 across lane halves, VGPR1 = K1|K3; B mirrored.
__device__ __forceinline__ v8f wmma_tile_k64(const float* __restrict__ A_lds,
                                             const float* __restrict__ W_glb,
                                             int mt, int nt, int lane) {
  const int l15  = lane & 15;
  const int half = lane >> 4;          // 0: K={0,1}, 1: K={2,3} of each 4-group
  const int m = mt * 16 + l15;
  const int n = nt * 16 + l15;
  v8f acc = {0.f, 0.f, 0.f, 0.f, 0.f, 0.f, 0.f, 0.f};
#pragma unroll
  for (int kk = 0; kk < 16; ++kk) {
    const int k0 = kk * 4 + half * 2;
    v2f a, b;
    a.x = A_lds[m * LDIM + k0];
    a.y = A_lds[m * LDIM + k0 + 1];
    b.x = W_glb[k0 * 64 + n];
    b.y = W_glb[(k0 + 1) * 64 + n];
    acc = __builtin_amdgcn_wmma_f32_16x16x4_f32(false, a, false, b,
                                                (short)0, acc, false, false);
  }
  return acc;
}

// -------- Kernel 1: per-batch spatial encode + temporal attention + GRU --------
// grid = B (32), block = 256 (8 wave32). Output: hidden_ws[b*64 + h].
__global__ __launch_bounds__(256)
void k1_spatial_att_gru(const float* __restrict__ x,
                        const float* __restrict__ W_s1, const float* __restrict__ b_s1,
                        const float* __restrict__ W_s2, const float* __restrict__ b_s2,
                        const float* __restrict__ Wa1,  const float* __restrict__ ba1,
                        const float* __restrict__ Wa2,  const float* __restrict__ ba2,
                        const float* __restrict__ W_ih, const float* __restrict__ b_ih,
                        const float* __restrict__ b_hh,
                        float* __restrict__ hidden_ws) {
  __shared__ float sA[96 * LDIM];   // h1, later tanh(h2@Wa1+ba1)
  __shared__ float sH2[96 * LDIM];  // h2
  __shared__ float sXb[96];
  __shared__ float sAtt[96];
  __shared__ float sNf[64];
  __shared__ float sGi[192];
  __shared__ float sRed[2];

  const int b = blockIdx.x;
  const int tid = threadIdx.x;
  const int lane = tid & 31;
  const int wave = tid >> 5;

  // xbar[t] = mean_n x[b,n,t]   (A_norm == ones/N collapses GCN aggregation)
  if (tid < 96) {
    float s = 0.f;
    const float* xp = x + (size_t)b * (128 * 96) + tid;
#pragma unroll 8
    for (int n = 0; n < 128; ++n) s += xp[n * 96];
    sXb[tid] = s * (1.0f / 128.0f);
  }
  __syncthreads();

  // h1[t,h] = relu(xbar[t]*W_s1[h] + b_s1[h])
  for (int i = tid; i < 96 * 64; i += 256) {
    const int t = i >> 6, h = i & 63;
    const float v = fmaf(sXb[t], W_s1[h], b_s1[h]);
    sA[t * LDIM + h] = v > 0.f ? v : 0.f;
  }
  __syncthreads();

  // h2 = relu(h1 @ W_s2 + b_s2) : 96x64x64, 24 tiles over 8 waves
#pragma unroll
  for (int ti = 0; ti < 3; ++ti) {
    const int tile = wave * 3 + ti;
    const int mt = tile >> 2, nt = tile & 3;
    v8f acc = wmma_tile_k64(sA, W_s2, mt, nt, lane);
    const int l15 = lane & 15, half = lane >> 4;
#pragma unroll
    for (int r = 0; r < 8; ++r) {
      const int mrow = mt * 16 + r + half * 8;
      const int ncol = nt * 16 + l15;
      const float v = acc[r] + b_s2[ncol];
      sH2[mrow * LDIM + ncol] = v > 0.f ? v : 0.f;
    }
  }
  __syncthreads();

  // tanh(h2 @ Wa1 + ba1) -> sA
#pragma unroll
  for (int ti = 0; ti < 3; ++ti) {
    const int tile = wave * 3 + ti;
    const int mt = tile >> 2, nt = tile & 3;
    v8f acc = wmma_tile_k64(sH2, Wa1, mt, nt, lane);
    const int l15 = lane & 15, half = lane >> 4;
#pragma unroll
    for (int r = 0; r < 8; ++r) {
      const int mrow = mt * 16 + r + half * 8;
      const int ncol = nt * 16 + l15;
      sA[mrow * LDIM + ncol] = tanhf(acc[r] + ba1[ncol]);
    }
  }
  __syncthreads();

  // attention logits + softmax over T=96
  if (tid < 96) {
    float s = ba2[0];
#pragma unroll 8
    for (int h = 0; h < 64; ++h) s = fmaf(sA[tid * LDIM + h], Wa2[h], s);
    sAtt[tid] = s;
  }
  __syncthreads();
  if (tid == 0) {
    float mx = sAtt[0];
    for (int t = 1; t < 96; ++t) mx = fmaxf(mx, sAtt[t]);
    float sum = 0.f;
    for (int t = 0; t < 96; ++t) sum += __expf(sAtt[t] - mx);
    sRed[0] = mx; sRed[1] = sum;
  }
  __syncthreads();
  if (tid < 96) sAtt[tid] = __expf(sAtt[tid] - sRed[0]) / sRed[1];
  __syncthreads();

  // nf[h] = sum_t h2[t,h] * w[t]
  if (tid < 64) {
    float s = 0.f;
    for (int t = 0; t < 96; ++t) s = fmaf(sH2[t * LDIM + tid], sAtt[t], s);
    sNf[tid] = s;
  }
  __syncthreads();

  // GRU input gates: gi = nf @ W_ih^T + b_ih   (192 dots of 64)
  if (tid < 192) {
    float s = b_ih[tid];
    const float* wr = W_ih + tid * 64;
#pragma unroll 8
    for (int h = 0; h < 64; ++h) s = fmaf(sNf[h], wr[h], s);
    sGi[tid] = s;
  }
  __syncthreads();

  // h0 == 0  =>  gh = b_hh; hidden = (1-z)*n
  if (tid < 64) {
    const float r  = sigmoidf_(sGi[tid] + b_hh[tid]);
    const float z  = sigmoidf_(sGi[64 + tid] + b_hh[64 + tid]);
    const float nn = tanhf(fmaf(r, b_hh[128 + tid], sGi[128 + tid]));
    hidden_ws[b * 64 + tid] = (1.f - z) * nn;
  }
}

// -------- Kernel 2: RK4(3/8) graph-ODE + output head + broadcast over N --------
__device__ __forceinline__ void gemm_tanh32(const float* __restrict__ in_lds,
                                            const float* __restrict__ W_glb,
                                            const float* __restrict__ bias_glb,
                                            float* __restrict__ out_lds,
                                            int wave, int lane) {
  const int mt = wave >> 2;  // 0..1  (32 rows)
  const int nt = wave & 3;   // 0..3  (64 cols)
  v8f acc = wmma_tile_k64(in_lds, W_glb, mt, nt, lane);
  const int l15 = lane & 15, half = lane >> 4;
#pragma unroll
  for (int r = 0; r < 8; ++r) {
    const int mrow = mt * 16 + r + half * 8;
    const int ncol = nt * 16 + l15;
    out_lds[mrow * LDIM + ncol] = tanhf(acc[r] + bias_glb[ncol]);
  }
}

// f(y) = tanh(tanh(y@W1+b1)@W2+b2); caller guarantees barrier before call.
__device__ __forceinline__ void feval(const float* __restrict__ in, float* __restrict__ out,
                                      float* __restrict__ mid,
                                      const float* __restrict__ W1, const float* __restrict__ b1,
                                      const float* __restrict__ W2, const float* __restrict__ b2,
                                      int wave, int lane) {
  gemm_tanh32(in, W1, b1, mid, wave, lane);
  __syncthreads();
  gemm_tanh32(mid, W2, b2, out, wave, lane);
  __syncthreads();
}

__global__ __launch_bounds__(256)
void k2_ode_head(const float* __restrict__ hidden_ws,
                 const float* __restrict__ W_o1, const float* __restrict__ b_o1,
                 const float* __restrict__ W_o2, const float* __restrict__ b_o2,
                 const float* __restrict__ W_out, const float* __restrict__ b_out,
                 float* __restrict__ out) {
  __shared__ float sY[32 * LDIM];
  __shared__ float sK1[32 * LDIM];
  __shared__ float sK2[32 * LDIM];
  __shared__ float sK3[32 * LDIM];
  __shared__ float sTin[32 * LDIM];   // stage input, then reused to hold k4
  __shared__ float sTmid[32 * LDIM];
  __shared__ float sPred[24 * 32];    // [t][b]

  const int tid = threadIdx.x;
  const int lane = tid & 31, wave = tid >> 5;
  const float dt = 24.0f / 23.0f;     // TF/(TF-1)

  for (int i = tid; i < 32 * 64; i += 256) {
    const int bb = i >> 6, h = i & 63;
    sY[bb * LDIM + h] = hidden_ws[i];
  }
  __syncthreads();

  if (tid < 32) {  // head at t=0
    float s = b_out[0];
#pragma unroll 8
    for (int h = 0; h < 64; ++h) s = fmaf(sY[tid * LDIM + h], W_out[h], s);
    sPred[tid] = s;
  }
  __syncthreads();

  for (int step = 1; step < 24; ++step) {
    feval(sY, sK1, sTmid, W_o1, b_o1, W_o2, b_o2, wave, lane);
    for (int i = tid; i < 32 * 64; i += 256) {
      const int p = (i >> 6) * LDIM + (i & 63);
      sTin[p] = fmaf(dt * (1.0f / 3.0f), sK1[p], sY[p]);
    }
    __syncthreads();
    feval(sTin, sK2, sTmid, W_o1, b_o1, W_o2, b_o2, wave, lane);
    for (int i = tid; i < 32 * 64; i += 256) {
      const int p = (i >> 6) * LDIM + (i & 63);
      sTin[p] = fmaf(dt, sK2[p] - (1.0f / 3.0f) * sK1[p], sY[p]);
    }
    __syncthreads();
    feval(sTin, sK3, sTmid, W_o1, b_o1, W_o2, b_o2, wave, lane);
    for (int i = tid; i < 32 * 64; i += 256) {
      const int p = (i >> 6) * LDIM + (i & 63);
      sTin[p] = fmaf(dt, sK1[p] - sK2[p] + sK3[p], sY[p]);
    }
    __syncthreads();
    feval(sTin, sTin, sTmid, W_o1, b_o1, W_o2, b_o2, wave, lane);  // k4 -> sTin
    for (int i = tid; i < 32 * 64; i += 256) {
      const int p = (i >> 6) * LDIM + (i & 63);
      sY[p] = fmaf(dt * 0.125f, sK1[p] + 3.0f * (sK2[p] + sK3[p]) + sTin[p], sY[p]);
    }
    __syncthreads();
    if (tid < 32) {
      float s = b_out[0];
#pragma unroll 8
      for (int h = 0; h < 64; ++h) s = fmaf(sY[tid * LDIM + h], W_out[h], s);
      sPred[step * 32 + tid] = s;
    }
    __syncthreads();
  }

  // out[b, n, t] = preds[t, b]  (constant over nodes), coalesced writes
  for (int i = tid; i < 32 * 128 * 24; i += 256) {
    const int bb = i / (128 * 24);
    const int t = i % 24;
    out[i] = sPred[t * 32 + bb];
  }
}

extern "C" void kernel_launch(void* const* d_in, const int* in_sizes, int n_in,
                              void* d_out, int out_size, void* d_ws, size_t ws_size,
                              hipStream_t stream) {
  (void)in_sizes; (void)n_in; (void)out_size; (void)ws_size;
  const float* x     = (const float*)d_in[0];
  const float* W_s1  = (const float*)d_in[1];
  const float* b_s1  = (const float*)d_in[2];
  const float* W_s2  = (const float*)d_in[3];
  const float* b_s2  = (const float*)d_in[4];
  const float* Wa1   = (const float*)d_in[5];
  const float* ba1   = (const float*)d_in[6];
  const float* Wa2   = (const float*)d_in[7];
  const float* ba2   = (const float*)d_in[8];
  const float* W_ih  = (const float*)d_in[9];
  // d_in[10] = W_hh: unused (h0 == 0 => gh = b_hh only)
  const float* b_ih  = (const float*)d_in[11];
  const float* b_hh  = (const float*)d_in[12];
  const float* W_o1  = (const float*)d_in[13];
  const float* b_o1  = (const float*)d_in[14];
  const float* W_o2  = (const float*)d_in[15];
  const float* b_o2  = (const float*)d_in[16];
  const float* W_out = (const float*)d_in[17];
  const float* b_out = (const float*)d_in[18];

  float* hidden_ws = (float*)d_ws;   // 32*64 floats
  float* out = (float*)d_out;        // 32*128*24 floats

  hipLaunchKernelGGL(k1_spatial_att_gru, dim3(32), dim3(256), 0, stream,
                     x, W_s1, b_s1, W_s2, b_s2, Wa1, ba1, Wa2, ba2,
                     W_ih, b_ih, b_hh, hidden_ws);
  hipLaunchKernelGGL(k2_ode_head, dim3(1), dim3(256), 0, stream,
                     hidden_ws, W_o1, b_o1, W_o2, b_o2, W_out, b_out, out);
}